// VocabParallelEmbeddingWithLoRA_81767587381865
// MI455X (gfx1250) — compile-verified
//
#include <hip/hip_runtime.h>
#include <hip/hip_bf16.h>

// Problem constants (match reference)
#define VOCAB 32000
#define EXTRA 256
#define DDIM  2048
#define RANK  16
#define NTOK  16384
#define VE    (VOCAB + EXTRA)   // 32256

typedef __attribute__((ext_vector_type(2))) float v2f;
typedef __attribute__((ext_vector_type(8))) float v8f;

// One block per 16-token row tile. 256 threads = 8 wave32; each wave covers
// 16 column tiles of 16 dims -> 8*16*16 = 2048 = D.
__global__ __launch_bounds__(256)
void lora_embed_kernel(const int* __restrict__ ids,
                       const int* __restrict__ widx,
                       const float* __restrict__ W,
                       const float* __restrict__ newemb,
                       const float* __restrict__ Abuf,
                       const float* __restrict__ Bbuf,
                       float* __restrict__ out)
{
    const int tileRow = blockIdx.x;          // 0..1023
    const int t0      = tileRow * 16;
    const int lane    = threadIdx.x & 31;
    const int wave    = threadIdx.x >> 5;    // 0..7
    const int half    = lane >> 4;           // 0 or 1 (lane group)
    const int lm      = lane & 15;

    __shared__ int s_ids[16];
    if (threadIdx.x < 16) s_ids[threadIdx.x] = ids[t0 + threadIdx.x];
    __syncthreads();

    // 1024 tokens per segment -> 64 row tiles share one LoRA slot
    const int lw = widx[tileRow >> 6];

    // ---- A fragments: 16x16 (M tokens x K ranks), four 16x4 K-slices.
    // Lane layout per ISA: lanes 0-15 -> M=lm, VGPR0/1 -> K = 4*ks + 2*half + {0,1}
    const int    myId = s_ids[lm];
    const float* Arow = Abuf + (size_t)lw * RANK * VE;
    v2f a[4];
#pragma unroll
    for (int ks = 0; ks < 4; ++ks) {
        const int r0 = 4 * ks + 2 * half;
        a[ks].x = Arow[(size_t)(r0    ) * VE + myId];
        a[ks].y = Arow[(size_t)(r0 + 1) * VE + myId];
    }

    const float* Bmat = Bbuf + (size_t)lw * DDIM * RANK;

#pragma unroll 4
    for (int j = 0; j < 16; ++j) {
        const int dbase = (wave * 16 + j) * 16;

        // ---- B fragments: B_wmma[k][n] = Bbuf[lw, dbase+n, k].
        // Lane n=lm loads float2 at r = 4*ks + 2*half  (8-byte aligned).
        const float* brow = Bmat + (size_t)(dbase + lm) * RANK + 2 * half;
        v2f b[4];
#pragma unroll
        for (int ks = 0; ks < 4; ++ks)
            b[ks] = *(const v2f*)(brow + 4 * ks);

        // ---- 16x16 f32 accumulator, 4x v_wmma_f32_16x16x4_f32 over K=16
        v8f c = {};
#pragma unroll
        for (int ks = 0; ks < 4; ++ks)
            c = __builtin_amdgcn_wmma_f32_16x16x4_f32(
                    /*neg_a=*/false, a[ks],
                    /*neg_b=*/false, b[ks],
                    /*c_mod=*/(short)0, c,
                    /*reuse_a=*/false, /*reuse_b=*/false);

        // ---- Epilogue: fuse base-embedding gather + add + store.
        // C/D layout: VGPR i -> row (8*half + i), col = lm.
#pragma unroll
        for (int i = 0; i < 8; ++i) {
            const int row = half * 8 + i;
            const int id  = s_ids[row];
            const float* basep;
            if (id < VOCAB) {
                basep = W + (size_t)id * DDIM;
            } else {
                basep = newemb + ((size_t)lw * EXTRA + (size_t)(id - VOCAB)) * DDIM;
            }
            const float v = basep[dbase + lm] + c[i];
            out[(size_t)(t0 + row) * DDIM + (size_t)(dbase + lm)] = v;
        }
    }
}

extern "C" void kernel_launch(void* const* d_in, const int* in_sizes, int n_in,
                              void* d_out, int out_size, void* d_ws, size_t ws_size,
                              hipStream_t stream) {
    const int*   ids    = (const int*)  d_in[0]; // input_ids            (T,)
    const int*   widx   = (const int*)  d_in[1]; // weight_indices       (BS,)
    const float* W      = (const float*)d_in[2]; // W                    (VOCAB, D)
    const float* newemb = (const float*)d_in[3]; // new_embeddings       (L, EXTRA, D)
    const float* Abuf   = (const float*)d_in[4]; // embedding_A          (L, R, VOCAB+EXTRA)
    const float* Bbuf   = (const float*)d_in[5]; // embedding_B          (L, D, R)
    float*       out    = (float*)d_out;         // (T, D) f32

    dim3 grid(NTOK / 16);   // 1024 row tiles
    dim3 block(256);        // 8 wave32
    lora_embed_kernel<<<grid, block, 0, stream>>>(ids, widx, W, newemb, Abuf, Bbuf, out);
}